// TemporalGraphModelGCNBatch_66477503807888
// MI455X (gfx1250) — compile-verified
//
#include <hip/hip_runtime.h>
#include <math.h>
#include <stdint.h>

#define EMB_D  128
#define GCN_H  128
#define LSTM_H 128
#define EDGE_A 16
#define T_SEQ  10

typedef __attribute__((ext_vector_type(2))) float v2f;
typedef __attribute__((ext_vector_type(8))) float v8f;
typedef int vsi4 __attribute__((vector_size(4 * sizeof(int))));  // matches builtin param

// ----------------------------------------------------------------------------
// CDNA5 fp32 WMMA: D(16x16) = A(16x4) * B(4x16) + C.
// Fragment layout per ISA 7.12.2 (wave32):
//   A: lane L -> M = L&15, K = 2*(L>>4) + vgpr   (2 VGPRs)
//   B: lane L -> N = L&15, K = 2*(L>>4) + vgpr   (2 VGPRs)
//   C/D: lane L -> N = L&15, M = 8*(L>>4) + vgpr (8 VGPRs)
// ----------------------------------------------------------------------------
__device__ __forceinline__ v8f wmma4(v2f a, v2f b, v8f c) {
  return __builtin_amdgcn_wmma_f32_16x16x4_f32(
      /*neg_a=*/false, a, /*neg_b=*/false, b,
      /*c_mod=*/(short)0, c, /*reuse_a=*/false, /*reuse_b=*/false);
}

__device__ __forceinline__ float sigm(float x) { return 1.0f / (1.0f + __expf(-x)); }

// ----------------------------------------------------------------------------
// Async global->LDS staging (gfx1250 GLOBAL_LOAD_ASYNC_TO_LDS_B128, ASYNCcnt),
// with a plain-copy fallback when the builtin is absent on this toolchain.
// ----------------------------------------------------------------------------
#ifndef __has_builtin
#define __has_builtin(x) 0
#endif
#if __has_builtin(__builtin_amdgcn_global_load_async_to_lds_b128)
#define HAVE_ASYNC_COPY 1
#else
#define HAVE_ASYNC_COPY 0
#endif

__device__ __forceinline__ void lds_copy16(const float* g, float* l) {
#if HAVE_ASYNC_COPY
  __builtin_amdgcn_global_load_async_to_lds_b128(
      (__attribute__((address_space(1))) vsi4*)(uintptr_t)g,
      (__attribute__((address_space(3))) vsi4*)(uintptr_t)l,
      /*offset=*/0, /*cpol=*/0);
#else
  *(float4*)l = *(const float4*)g;
#endif
}

__device__ __forceinline__ void lds_copy_wait() {
#if HAVE_ASYNC_COPY
#if __has_builtin(__builtin_amdgcn_s_wait_asynccnt)
  __builtin_amdgcn_s_wait_asynccnt(0);
#else
  asm volatile("s_wait_asynccnt 0x0" ::: "memory");
#endif
#endif
}

// ----------------------------------------------------------------------------
// GCN normalization: deg (with self loop) -> dinv
// ----------------------------------------------------------------------------
__global__ void k_deg_init(float* __restrict__ deg, int N) {
  int i = blockIdx.x * blockDim.x + threadIdx.x;
  if (i < N) deg[i] = 1.0f;  // self-loop
}

__global__ void k_deg_accum(float* __restrict__ deg, const int* __restrict__ dst, int E) {
  int e = blockIdx.x * blockDim.x + threadIdx.x;
  if (e < E) atomicAdd(&deg[dst[e]], 1.0f);
}

__global__ void k_dinv(float* __restrict__ deg, int N) {
  int i = blockIdx.x * blockDim.x + threadIdx.x;
  if (i < N) {
    float d = deg[i];
    deg[i] = (d > 0.0f) ? rsqrtf(d) : 0.0f;  // in place: now holds deg^-1/2
  }
}

// ----------------------------------------------------------------------------
// xw = emb @ gcn_w, fp32 WMMA. One wave -> 32 rows x 128 cols.
// gcn_w async-staged into LDS, then repacked into K-paired layout so every
// B-fragment is one aligned ds_load_b64 (pair stride padded: conflict-free).
// ----------------------------------------------------------------------------
#define W2S 144  // pairs per k2-row: 128 + 16 pad (bank spread)

__global__ __launch_bounds__(256) void k_xw(const float* __restrict__ emb,
                                            const float* __restrict__ w,
                                            float* __restrict__ xw, int N) {
  __shared__ __align__(16) float wraw[EMB_D * GCN_H];  // 64 KB linear image
  __shared__ __align__(16) float wl2[64 * W2S * 2];    // 72 KB paired image

  // stage 1: linear copy (async if available)
  for (int c = threadIdx.x; c < (EMB_D * GCN_H) / 4; c += 256)
    lds_copy16(w + 4 * c, wraw + 4 * c);
  lds_copy_wait();
  __syncthreads();

  // stage 2: repack into pairs {W[2k2][n], W[2k2+1][n]}
  for (int i = threadIdx.x; i < EMB_D * GCN_H; i += 256) {
    int k = i >> 7, n = i & 127;
    wl2[((k >> 1) * W2S + n) * 2 + (k & 1)] = wraw[i];
  }
  __syncthreads();

  const int wave = threadIdx.x >> 5, lane = threadIdx.x & 31;
  const int rows = (blockIdx.x * 8 + wave) * 32;
  if (rows >= N) return;  // wave-uniform branch: EXEC all-ones for WMMA

  const int mn = lane & 15;        // M (A) / N (B,C)
  const int kh = (lane >> 4) * 2;  // K pair chosen by half-wave
  const float* a0 = emb + (size_t)(rows + mn) * EMB_D;
  const float* a1 = emb + (size_t)(rows + 16 + mn) * EMB_D;

  v8f acc[2][8];
  for (int r = 0; r < 2; ++r)
    for (int j = 0; j < 8; ++j)
      for (int v = 0; v < 8; ++v) acc[r][j][v] = 0.0f;

  #pragma unroll 2
  for (int ks = 0; ks < EMB_D; ks += 4) {
    const v2f A0 = *(const v2f*)&a0[ks + kh];
    const v2f A1 = *(const v2f*)&a1[ks + kh];
    const int k2 = (ks >> 1) + (lane >> 4);
    #pragma unroll
    for (int j = 0; j < 8; ++j) {
      const v2f b = *(const v2f*)&wl2[(k2 * W2S + j * 16 + mn) * 2];
      acc[0][j] = wmma4(A0, b, acc[0][j]);
      acc[1][j] = wmma4(A1, b, acc[1][j]);
    }
  }

  const int mb = (lane >> 4) * 8;
  for (int r = 0; r < 2; ++r)
    for (int j = 0; j < 8; ++j)
      for (int v = 0; v < 8; ++v)
        xw[(size_t)(rows + r * 16 + mb + v) * GCN_H + j * 16 + mn] = acc[r][j][v];
}

// ----------------------------------------------------------------------------
// gcn_out init: self-loop message + bias:  out[i] = xw[i]*dinv[i]^2 + b
// ----------------------------------------------------------------------------
__global__ void k_gcn_init(const float* __restrict__ xw, const float* __restrict__ dinv,
                           const float* __restrict__ b, float* __restrict__ gcn, int N) {
  long long i = (long long)blockIdx.x * blockDim.x + threadIdx.x;
  if (i < (long long)N * GCN_H) {
    int node = (int)(i >> 7), c = (int)(i & (GCN_H - 1));
    float dv = dinv[node];
    gcn[i] = xw[i] * dv * dv + b[c];
  }
}

// ----------------------------------------------------------------------------
// Edge scatter: gcn[dst] += xw[src] * dinv[src]*dinv[dst]
// One wave per edge; 4 floats per lane; L2-resident f32 atomics.
// ----------------------------------------------------------------------------
__global__ __launch_bounds__(256) void k_scatter(const float* __restrict__ xw,
                                                 const float* __restrict__ dinv,
                                                 const int* __restrict__ src,
                                                 const int* __restrict__ dst,
                                                 float* __restrict__ gcn, int E) {
  const int wave = threadIdx.x >> 5, lane = threadIdx.x & 31;
  const int e = blockIdx.x * 8 + wave;
  if (e >= E) return;
  const int s = src[e], d = dst[e];
  const float norm = dinv[s] * dinv[d];
  const float4 v = ((const float4*)(xw + (size_t)s * GCN_H))[lane];
  float* o = gcn + (size_t)d * GCN_H + lane * 4;
  atomicAdd(o + 0, v.x * norm);
  atomicAdd(o + 1, v.y * norm);
  atomicAdd(o + 2, v.z * norm);
  atomicAdd(o + 3, v.w * norm);
}

// ----------------------------------------------------------------------------
// LSTM over T=10, gates i,f,g,o. One wave owns 16 batch rows.
// c,h live in registers (C-matrix layout). h round-trips through a padded
// (stride-132, bank-conflict-free) LDS tile each step to become the
// A-fragment of the recurrent matmul; A-fragments are hoisted into registers
// in 64-k chunks and reused across all four gates.
// ----------------------------------------------------------------------------
#define HLS (LSTM_H + 4)  // 132 floats per row: conflict-free A-frag reads

__global__ __launch_bounds__(256) void k_lstm(const float* __restrict__ x,      // [B,T,16]
                                              const float* __restrict__ w_ih,   // [512,16]
                                              const float* __restrict__ w_hh,   // [512,128]
                                              const float* __restrict__ b_ih,
                                              const float* __restrict__ b_hh,
                                              float* __restrict__ hout, int B) { // [B,128]
  __shared__ __align__(16) float hl[8][16 * HLS];  // 8 waves * 8.25 KB
  const int wave = threadIdx.x >> 5, lane = threadIdx.x & 31;
  const int strip0 = blockIdx.x * 8 + wave;
  const bool active = (strip0 * 16 < B);
  const int strip = active ? strip0 : 0;  // inactive waves shadow strip 0, skip store
  const int mn = lane & 15;
  const int kh = (lane >> 4) * 2;
  const int mb = (lane >> 4) * 8;
  float* myh = hl[wave];

  v8f c[8], h[8];
  for (int j = 0; j < 8; ++j)
    for (int v = 0; v < 8; ++v) { c[j][v] = 0.0f; h[j][v] = 0.0f; }

  #pragma unroll 1
  for (int t = 0; t < T_SEQ; ++t) {
    // publish h in row-major [16 x HLS] for A-fragment reads
    #pragma unroll
    for (int j = 0; j < 8; ++j)
      #pragma unroll
      for (int v = 0; v < 8; ++v)
        myh[(mb + v) * HLS + j * 16 + mn] = h[j][v];
    __syncthreads();

    // x_t A-fragments (K = 16 -> 4 k-steps), preloaded as aligned pairs
    const float* xr = x + ((size_t)(strip * 16 + mn) * T_SEQ + t) * EDGE_A;
    v2f ax[4];
    #pragma unroll
    for (int ks = 0; ks < 4; ++ks) ax[ks] = *(const v2f*)&xr[ks * 4 + kh];

    #pragma unroll 1
    for (int j = 0; j < 8; ++j) {   // 16-column block of the hidden dim
      v8f g[4];
      #pragma unroll
      for (int q = 0; q < 4; ++q) { // gates i,f,g,o
        const int col = q * LSTM_H + j * 16 + mn;  // row of w_ih / w_hh
        const float bb = b_ih[col] + b_hh[col];
        #pragma unroll
        for (int v = 0; v < 8; ++v) g[q][v] = bb;
        // x_t @ w_ih^T : B[k][n] = w_ih[col][k], contiguous pair -> b64
        #pragma unroll
        for (int ks = 0; ks < 4; ++ks) {
          const v2f b = *(const v2f*)&w_ih[col * EDGE_A + ks * 4 + kh];
          g[q] = wmma4(ax[ks], b, g[q]);
        }
      }
      // h @ w_hh^T : A-fragments hoisted per 64-k chunk, shared by all gates
      #pragma unroll 1
      for (int half = 0; half < 2; ++half) {
        v2f ah[16];
        #pragma unroll
        for (int ks = 0; ks < 16; ++ks)
          ah[ks] = *(const v2f*)&myh[mn * HLS + (half * 16 + ks) * 4 + kh];
        #pragma unroll
        for (int q = 0; q < 4; ++q) {
          const int col = q * LSTM_H + j * 16 + mn;
          #pragma unroll
          for (int ks = 0; ks < 16; ++ks) {
            const v2f b = *(const v2f*)&w_hh[col * LSTM_H + (half * 16 + ks) * 4 + kh];
            g[q] = wmma4(ah[ks], b, g[q]);
          }
        }
      }
      // elementwise LSTM cell update, directly in C-matrix layout
      #pragma unroll
      for (int v = 0; v < 8; ++v) {
        float ig = sigm(g[0][v]);
        float fg = sigm(g[1][v]);
        float gg = tanhf(g[2][v]);
        float og = sigm(g[3][v]);
        float cv = fg * c[j][v] + ig * gg;
        c[j][v] = cv;
        h[j][v] = og * tanhf(cv);
      }
    }
    __syncthreads();  // before next step rewrites LDS h tiles
  }

  if (active) {
    for (int j = 0; j < 8; ++j)
      for (int v = 0; v < 8; ++v)
        hout[(size_t)(strip * 16 + mb + v) * LSTM_H + j * 16 + mn] = h[j][v];
  }
}

// ----------------------------------------------------------------------------
// out[b] = [gcn[sender[b]], gcn[receiver[b]], h[b]] . pred_w + pred_b
// One wave per b, coalesced row reads, shuffle reduction.
// ----------------------------------------------------------------------------
__global__ __launch_bounds__(256) void k_pred(const int* __restrict__ sender,
                                              const int* __restrict__ receiver,
                                              const float* __restrict__ gcn,
                                              const float* __restrict__ hl,
                                              const float* __restrict__ pw,
                                              const float* __restrict__ pb,
                                              float* __restrict__ out, int B) {
  const int wave = threadIdx.x >> 5, lane = threadIdx.x & 31;
  const int b = blockIdx.x * 8 + wave;
  if (b >= B) return;
  const float* s = gcn + (size_t)sender[b] * GCN_H;
  const float* r = gcn + (size_t)receiver[b] * GCN_H;
  const float* e = hl + (size_t)b * LSTM_H;
  float acc = 0.0f;
  #pragma unroll
  for (int c0 = 0; c0 < GCN_H; c0 += 32) {
    const int c = c0 + lane;
    acc += s[c] * pw[c] + r[c] * pw[GCN_H + c] + e[c] * pw[2 * GCN_H + c];
  }
  #pragma unroll
  for (int off = 16; off > 0; off >>= 1) acc += __shfl_xor(acc, off, 32);
  if (lane == 0) out[b] = acc + pb[0];
}

// ----------------------------------------------------------------------------
extern "C" void kernel_launch(void* const* d_in, const int* in_sizes, int n_in,
                              void* d_out, int out_size, void* d_ws, size_t ws_size,
                              hipStream_t stream) {
  const int*   sender        = (const int*)  d_in[0];
  const int*   receiver      = (const int*)  d_in[1];
  const float* edge_features = (const float*)d_in[2];
  const int*   edge_index    = (const int*)  d_in[3];
  const float* emb           = (const float*)d_in[4];
  const float* gcn_w         = (const float*)d_in[5];
  const float* gcn_b         = (const float*)d_in[6];
  const float* w_ih          = (const float*)d_in[7];
  const float* w_hh          = (const float*)d_in[8];
  const float* b_ih          = (const float*)d_in[9];
  const float* b_hh          = (const float*)d_in[10];
  const float* pred_w        = (const float*)d_in[11];
  const float* pred_b        = (const float*)d_in[12];
  float* out = (float*)d_out;

  const int B = in_sizes[0];
  const int E = in_sizes[3] / 2;
  const int N = in_sizes[4] / EMB_D;
  const int* src = edge_index;
  const int* dst = edge_index + E;

  // workspace carve (deg/dinv, xw, gcn_out, lstm_h) ~ 120 MB
  char* p = (char*)d_ws;
  auto carve = [&](size_t bytes) -> float* {
    float* r = (float*)p;
    p += (bytes + 255) & ~(size_t)255;
    return r;
  };
  float* deg = carve((size_t)N * sizeof(float));
  float* xw  = carve((size_t)N * GCN_H * sizeof(float));
  float* gcn = carve((size_t)N * GCN_H * sizeof(float));
  float* hl  = carve((size_t)B * LSTM_H * sizeof(float));
  (void)ws_size; (void)n_in; (void)out_size;

  k_deg_init <<<(N + 255) / 256, 256, 0, stream>>>(deg, N);
  k_deg_accum<<<(E + 255) / 256, 256, 0, stream>>>(deg, dst, E);
  k_dinv     <<<(N + 255) / 256, 256, 0, stream>>>(deg, N);

  const int strips32 = (N + 31) / 32;
  k_xw       <<<(strips32 + 7) / 8, 256, 0, stream>>>(emb, gcn_w, xw, N);

  const long long ne = (long long)N * GCN_H;
  k_gcn_init <<<(unsigned)((ne + 255) / 256), 256, 0, stream>>>(xw, deg, gcn_b, gcn, N);
  k_scatter  <<<(E + 7) / 8, 256, 0, stream>>>(xw, deg, src, dst, gcn, E);

  const int bstrips = (B + 15) / 16;
  k_lstm     <<<(bstrips + 7) / 8, 256, 0, stream>>>(edge_features, w_ih, w_hh, b_ih, b_hh, hl, B);

  k_pred     <<<(B + 7) / 8, 256, 0, stream>>>(sender, receiver, gcn, hl, pred_w, pred_b, out, B);
}